// PointNetSetAbstraction_67757404062295
// MI455X (gfx1250) — compile-verified
//
#include <hip/hip_runtime.h>
#include <hip/hip_bf16.h>

typedef __attribute__((ext_vector_type(2))) float v2f;
typedef __attribute__((ext_vector_type(8))) float v8f;

#define NPTS   16384
#define NPOINT 512
#define NSAMP  32
#define CIN    64      // 3 xyz + 61 point features
#define NCOLS  2048    // NSAMP * CIN
#define NEGF   (-3.4028234663852886e38f)

// ---------------------------------------------------------------------------
// Kernel 1: farthest point sampling. One block per batch, 1024 threads,
// 16 points per thread held in registers. Writes new_xyz (B,512,3) to d_out.
// ---------------------------------------------------------------------------
__global__ __launch_bounds__(1024) void fps_kernel(const float* __restrict__ xyz,
                                                   float* __restrict__ new_xyz) {
  const int b = blockIdx.x;
  const int t = threadIdx.x;
  const float* xb = xyz + (size_t)b * NPTS * 3;

  float px[16], py[16], pz[16], dist[16];
#pragma unroll
  for (int j = 0; j < 16; ++j) {
    int p = t + 1024 * j;
    px[j] = xb[p * 3 + 0];
    py[j] = xb[p * 3 + 1];
    pz[j] = xb[p * 3 + 2];
    dist[j] = 1e10f;
  }

  __shared__ float cs[3];
  __shared__ float wval[32];
  __shared__ int widx[32];
  __shared__ int sfar;

  int far = 0;
  for (int it = 0; it < NPOINT; ++it) {
    // owner of the current farthest point publishes its coords + output
    if (t == (far & 1023)) {
      int j = far >> 10;
      cs[0] = px[j]; cs[1] = py[j]; cs[2] = pz[j];
      float* o = new_xyz + ((size_t)b * NPOINT + it) * 3;
      o[0] = px[j]; o[1] = py[j]; o[2] = pz[j];
    }
    __syncthreads();
    float cx = cs[0], cy = cs[1], cz = cs[2];

    float bestv = -1.0f;
    int besti = 0;
#pragma unroll
    for (int j = 0; j < 16; ++j) {
      float dx = px[j] - cx, dy = py[j] - cy, dz = pz[j] - cz;
      float d = dx * dx + dy * dy + dz * dz;
      dist[j] = fminf(dist[j], d);
      if (dist[j] > bestv) { bestv = dist[j]; besti = t + 1024 * j; }
    }
    // wave32 reduction (max value, ties -> lowest index)
#pragma unroll
    for (int off = 16; off; off >>= 1) {
      float ov = __shfl_xor(bestv, off);
      int oi = __shfl_xor(besti, off);
      if (ov > bestv || (ov == bestv && oi < besti)) { bestv = ov; besti = oi; }
    }
    if ((t & 31) == 0) { wval[t >> 5] = bestv; widx[t >> 5] = besti; }
    __syncthreads();
    if (t < 32) {
      float v = wval[t];
      int i = widx[t];
#pragma unroll
      for (int off = 16; off; off >>= 1) {
        float ov = __shfl_xor(v, off);
        int oi = __shfl_xor(i, off);
        if (ov > v || (ov == v && oi < i)) { v = ov; i = oi; }
      }
      if (t == 0) sfar = i;
    }
    __syncthreads();
    far = sfar;
  }
}

// ---------------------------------------------------------------------------
// Kernel 2: ball query + grouping. One wave per centroid (8 waves/block).
// Builds x0[b][s][k*64+c] : c<3 -> xyz[idx]-centroid, else points[idx][c-3].
// ---------------------------------------------------------------------------
__global__ __launch_bounds__(256) void ballquery_kernel(const float* __restrict__ xyz,
                                                        const float* __restrict__ pts,
                                                        const float* __restrict__ new_xyz,
                                                        float* __restrict__ x0) {
  const int wave = threadIdx.x >> 5;
  const int lane = threadIdx.x & 31;
  const int s = blockIdx.x * 8 + wave;
  const int b = blockIdx.y;
  const float* xb = xyz + (size_t)b * NPTS * 3;

  const float* cp = new_xyz + ((size_t)b * NPOINT + s) * 3;
  const float cx = cp[0], cy = cp[1], cz = cp[2];
  const float r2 = 0.2f * 0.2f;

  __shared__ int sidx[8][NSAMP];

  int count = 0;
  for (int q = 0; q < NPTS && count < NSAMP; q += 32) {
    int p = q + lane;
    float dx = xb[p * 3 + 0] - cx;
    float dy = xb[p * 3 + 1] - cy;
    float dz = xb[p * 3 + 2] - cz;
    bool in = (dx * dx + dy * dy + dz * dz) <= r2;
    unsigned m = (unsigned)__ballot(in);
    int pos = count + __popc(m & ((1u << lane) - 1u));
    if (in && pos < NSAMP) sidx[wave][pos] = p;
    count += __popc(m);
  }
  if (count > NSAMP) count = NSAMP;
  __syncthreads();
  // pad with first in-radius index (centroid itself guarantees count >= 1)
  int first = sidx[wave][0];
  for (int pos = count + lane; pos < NSAMP; pos += 32) sidx[wave][pos] = first;
  __syncthreads();

  float* orow0 = x0 + ((size_t)(b * NPOINT + s)) * NCOLS;
  for (int k = 0; k < NSAMP; ++k) {
    int ik = sidx[wave][k];
    const float* prow = pts + ((size_t)b * NPTS + ik) * 61;
    const float* xrow = xyz + ((size_t)b * NPTS + ik) * 3;
    float* orow = orow0 + k * CIN;
#pragma unroll
    for (int c0 = 0; c0 < 2; ++c0) {
      int c = lane + 32 * c0;
      float v;
      if (c < 3) {
        float cc = (c == 0) ? cx : ((c == 1) ? cy : cz);
        v = xrow[c] - cc;
      } else {
        v = prow[c - 3];
      }
      orow[c] = v;
    }
  }
}

// ---------------------------------------------------------------------------
// Kernel 3: init pooled output to float32 min (pool identity).
// ---------------------------------------------------------------------------
__global__ void init_pooled_kernel(float* __restrict__ p, int n) {
  int i = blockIdx.x * 256 + threadIdx.x;
  if (i < n) p[i] = NEGF;
}

__device__ inline void atomicMaxF32(float* addr, float val) {
  unsigned* a = (unsigned*)addr;
  unsigned old = *a;
  while (true) {
    float f = __uint_as_float(old);
    if (f >= val) break;
    unsigned assumed = old;
    old = atomicCAS(a, assumed, __float_as_uint(val));
    if (old == assumed) break;
  }
}

// ---------------------------------------------------------------------------
// WMMA f32 GEMM:  Y[b] = relu(bn(W (MxK) @ X[b] (KxN)))  with K=512, N=2048.
// Block: 256 threads / 8 waves, tile 64(M) x 128(N), K-step 32 via LDS.
// Each wave: 32x32 patch = 2x2 tiles of v_wmma_f32_16x16x4_f32.
// POOL variant fuses masked max-pool over k = n>>6 via atomic max.
// ---------------------------------------------------------------------------
template <bool POOL>
__global__ __launch_bounds__(256) void wmma_gemm_kernel(
    const float* __restrict__ W, const float* __restrict__ X,
    const float* __restrict__ bi, const float* __restrict__ gg,
    const float* __restrict__ be, const float* __restrict__ mm,
    const float* __restrict__ vv, float* __restrict__ Y,
    const float* __restrict__ mask, float* __restrict__ pooled, int M) {
  const int K = 512, N = NCOLS;
  __shared__ float As[64][36];   // stride 36: conflict-free column reads
  __shared__ float Bs[32][132];

  const int b = blockIdx.z;
  const float* Xb = X + (size_t)b * K * N;
  const int tid = threadIdx.x;
  const int wv = tid >> 5, lane = tid & 31;
  const int hf = lane >> 4, lr = lane & 15;
  const int wm = (wv >> 2) * 32, wn = (wv & 3) * 32;
  const int blkM = blockIdx.x * 64, blkN = blockIdx.y * 128;

  v8f acc[2][2] = {};

  for (int kb = 0; kb < K; kb += 32) {
#pragma unroll
    for (int i = 0; i < 2; ++i) {   // A tile: 64x32 = 512 float4
      int f = tid + i * 256;
      int r = f >> 3, c4 = (f & 7) << 2;
      const float4 aa = *(const float4*)(W + (size_t)(blkM + r) * K + kb + c4);
      As[r][c4 + 0] = aa.x; As[r][c4 + 1] = aa.y;
      As[r][c4 + 2] = aa.z; As[r][c4 + 3] = aa.w;
    }
#pragma unroll
    for (int i = 0; i < 4; ++i) {   // B tile: 32x128 = 1024 float4
      int f = tid + i * 256;
      int r = f >> 5, c4 = (f & 31) << 2;
      const float4 bb = *(const float4*)(Xb + (size_t)(kb + r) * N + blkN + c4);
      Bs[r][c4 + 0] = bb.x; Bs[r][c4 + 1] = bb.y;
      Bs[r][c4 + 2] = bb.z; Bs[r][c4 + 3] = bb.w;
    }
    __syncthreads();

#pragma unroll
    for (int k0 = 0; k0 < 32; k0 += 4) {
      v2f a[2], bf[2];
#pragma unroll
      for (int tm = 0; tm < 2; ++tm) {      // A frag: lanes 0-15 K=0,1; 16-31 K=2,3
        int row = wm + tm * 16 + lr;
        a[tm].x = As[row][k0 + 2 * hf];
        a[tm].y = As[row][k0 + 2 * hf + 1];
      }
#pragma unroll
      for (int tn = 0; tn < 2; ++tn) {      // B frag mirrored
        int col = wn + tn * 16 + lr;
        bf[tn].x = Bs[k0 + 2 * hf][col];
        bf[tn].y = Bs[k0 + 2 * hf + 1][col];
      }
#pragma unroll
      for (int tm = 0; tm < 2; ++tm)
#pragma unroll
        for (int tn = 0; tn < 2; ++tn)
          acc[tm][tn] = __builtin_amdgcn_wmma_f32_16x16x4_f32(
              false, a[tm], false, bf[tn], (short)0, acc[tm][tn], false, false);
    }
    __syncthreads();
  }

  // epilogue: bias + batchnorm + relu (+ masked max-pool)
#pragma unroll
  for (int tm = 0; tm < 2; ++tm) {
#pragma unroll
    for (int r = 0; r < 8; ++r) {
      int m = blkM + wm + tm * 16 + hf * 8 + r;   // C/D layout: rows r / r+8
      float sc = gg[m] * rsqrtf(vv[m] + 1e-5f);
      float sh = (bi[m] - mm[m]) * sc + be[m];
#pragma unroll
      for (int tn = 0; tn < 2; ++tn) {
        int n = blkN + wn + tn * 16 + lr;
        float y = acc[tm][tn][r] * sc + sh;
        y = fmaxf(y, 0.0f);
        if (!POOL) {
          Y[((size_t)b * M + m) * N + n] = y;
        } else {
          int k = n >> 6, c = n & 63;
          float mk = mask[b * NSAMP + k];
          float val = y * mk + (1.0f - mk) * NEGF;
          atomicMaxF32(pooled + ((size_t)b * M + m) * CIN + c, val);
        }
      }
    }
  }
}

// ---------------------------------------------------------------------------
extern "C" void kernel_launch(void* const* d_in, const int* in_sizes, int n_in,
                              void* d_out, int out_size, void* d_ws, size_t ws_size,
                              hipStream_t stream) {
  (void)in_sizes; (void)n_in; (void)out_size; (void)ws_size;
  const float* xyz  = (const float*)d_in[0];
  const float* pts  = (const float*)d_in[1];
  const float* mask = (const float*)d_in[2];
  const float* w1 = (const float*)d_in[3];
  const float* b1 = (const float*)d_in[4];
  const float* g1 = (const float*)d_in[5];
  const float* be1 = (const float*)d_in[6];
  const float* m1 = (const float*)d_in[7];
  const float* v1 = (const float*)d_in[8];
  const float* w2 = (const float*)d_in[9];
  const float* b2 = (const float*)d_in[10];
  const float* g2 = (const float*)d_in[11];
  const float* be2 = (const float*)d_in[12];
  const float* m2 = (const float*)d_in[13];
  const float* v2 = (const float*)d_in[14];

  float* new_xyz = (float*)d_out;                       // (8,512,3)
  float* pooled = (float*)d_out + 8 * NPOINT * 3;       // (8,1024,64)

  float* x0 = (float*)d_ws;                             // (8,512,2048)
  float* x1 = x0 + (size_t)8 * NPOINT * NCOLS;          // (8,512,2048)

  fps_kernel<<<8, 1024, 0, stream>>>(xyz, new_xyz);
  ballquery_kernel<<<dim3(NPOINT / 8, 8), 256, 0, stream>>>(xyz, pts, new_xyz, x0);

  int npool = 8 * 1024 * CIN;
  init_pooled_kernel<<<(npool + 255) / 256, 256, 0, stream>>>(pooled, npool);

  wmma_gemm_kernel<false><<<dim3(512 / 64, NCOLS / 128, 8), 256, 0, stream>>>(
      w1, x0, b1, g1, be1, m1, v1, x1, nullptr, nullptr, 512);
  wmma_gemm_kernel<true><<<dim3(1024 / 64, NCOLS / 128, 8), 256, 0, stream>>>(
      w2, x1, b2, g2, be2, m2, v2, nullptr, mask, pooled, 1024);
}